// GATRanker_54073638257074
// MI455X (gfx1250) — compile-verified
//
#include <hip/hip_runtime.h>
#include <hip/hip_bf16.h>

#define IN_CH 128
#define HID   64
#define NEG_SLOPE 0.2f

typedef __attribute__((ext_vector_type(2))) float v2f;
typedef __attribute__((ext_vector_type(8))) float v8f;

// ---------------------------------------------------------------------------
// Monotone uint encoding of float for race-free atomicMax over signed floats
// ---------------------------------------------------------------------------
__device__ __forceinline__ unsigned fkey(float f) {
    unsigned b = __float_as_uint(f);
    return (b & 0x80000000u) ? ~b : (b | 0x80000000u);
}
__device__ __forceinline__ float funkey(unsigned k) {
    unsigned b = (k & 0x80000000u) ? (k & 0x7FFFFFFFu) : ~k;
    return __uint_as_float(b);
}

// ---------------------------------------------------------------------------
// init: out = bias (broadcast), m_bits = key(-inf-ish)=0, s = 0
// ---------------------------------------------------------------------------
__global__ void __launch_bounds__(256)
gat_init(float* __restrict__ out, const float* __restrict__ bias,
         unsigned* __restrict__ m_bits, float* __restrict__ s, int N) {
    int i = blockIdx.x * blockDim.x + threadIdx.x;
    if (i < N * HID) out[i] = bias[i & (HID - 1)];
    if (i < N) { m_bits[i] = 0u; s[i] = 0.0f; }
}

// ---------------------------------------------------------------------------
// GEMM h = x @ W via V_WMMA_F32_16X16X4_F32.
// One wave computes a 16x64 slab of h (4 accumulators of 16x16), K=128 in
// 32 steps of 4. W staged in LDS (32 KB). 8 waves / block -> 128 rows/block.
//
// f32 A 16x4 layout : lanes 0-15 K={0,1}, lanes 16-31 K={2,3}  (M = lane&15)
// f32 B 4x16 layout : VGPR0 = rows K0/K2, VGPR1 = rows K1/K3  (N = lane&15)
// f32 C 16x16       : VGPR v -> M = v + 8*(lane>=16), N = lane&15
// ---------------------------------------------------------------------------
__global__ void __launch_bounds__(256)
gat_gemm_wmma(const float* __restrict__ x, const float* __restrict__ W,
              float* __restrict__ h, int N) {
    __shared__ float Wlds[IN_CH * HID];  // 32 KB
    for (int i = threadIdx.x; i < IN_CH * HID; i += 256)
        Wlds[i] = W[i];
    __syncthreads();

    const int wave = threadIdx.x >> 5;
    const int lane = threadIdx.x & 31;
    const int half = lane >> 4;     // 0: lanes 0-15, 1: lanes 16-31
    const int l    = lane & 15;

    const int r0 = (blockIdx.x * 8 + wave) * 16;
    if (r0 >= N) return;  // wave-uniform: EXEC stays all-ones for WMMA

    const int arow = min(r0 + l, N - 1);            // clamped A row (full EXEC)
    const float* __restrict__ xrow = x + (size_t)arow * IN_CH;

    v8f acc0 = {}, acc1 = {}, acc2 = {}, acc3 = {};

    #pragma unroll 4
    for (int kk = 0; kk < IN_CH; kk += 4) {
        const int ka = kk + 2 * half;
        v2f a;
        a.x = xrow[ka];
        a.y = xrow[ka + 1];
        const float* w0 = &Wlds[ka * HID];
        const float* w1 = &Wlds[(ka + 1) * HID];
        v2f b0, b1, b2, b3;
        b0.x = w0[l];      b0.y = w1[l];
        b1.x = w0[16 + l]; b1.y = w1[16 + l];
        b2.x = w0[32 + l]; b2.y = w1[32 + l];
        b3.x = w0[48 + l]; b3.y = w1[48 + l];
        acc0 = __builtin_amdgcn_wmma_f32_16x16x4_f32(false, a, false, b0, (short)0, acc0, false, false);
        acc1 = __builtin_amdgcn_wmma_f32_16x16x4_f32(false, a, false, b1, (short)0, acc1, false, false);
        acc2 = __builtin_amdgcn_wmma_f32_16x16x4_f32(false, a, false, b2, (short)0, acc2, false, false);
        acc3 = __builtin_amdgcn_wmma_f32_16x16x4_f32(false, a, false, b3, (short)0, acc3, false, false);
    }

    #pragma unroll
    for (int v = 0; v < 8; ++v) {
        int row = r0 + v + 8 * half;
        if (row < N) {
            float* hr = h + (size_t)row * HID;
            hr[ 0 + l] = acc0[v];
            hr[16 + l] = acc1[v];
            hr[32 + l] = acc2[v];
            hr[48 + l] = acc3[v];
        }
    }
}

// ---------------------------------------------------------------------------
// a_src[n] = h[n,:] . att_src ; a_dst[n] = h[n,:] . att_dst
// ---------------------------------------------------------------------------
__global__ void __launch_bounds__(256)
gat_att_dots(const float* __restrict__ h, const float* __restrict__ att_src,
             const float* __restrict__ att_dst, float* __restrict__ a_src,
             float* __restrict__ a_dst, int N) {
    int n = blockIdx.x * blockDim.x + threadIdx.x;
    if (n >= N) return;
    const float4* hr = (const float4*)(h + (size_t)n * HID);
    float s0 = 0.0f, s1 = 0.0f;
    #pragma unroll
    for (int q = 0; q < HID / 4; ++q) {
        float4 v = hr[q];
        s0 += v.x * att_src[4*q+0] + v.y * att_src[4*q+1] + v.z * att_src[4*q+2] + v.w * att_src[4*q+3];
        s1 += v.x * att_dst[4*q+0] + v.y * att_dst[4*q+1] + v.z * att_dst[4*q+2] + v.w * att_dst[4*q+3];
    }
    a_src[n] = s0;
    a_dst[n] = s1;
}

// ---------------------------------------------------------------------------
// Pass A: raw leaky-relu score per edge -> alpha slot; segment-max via
// monotone-uint atomicMax (deterministic result).
// ---------------------------------------------------------------------------
__global__ void __launch_bounds__(256)
gat_score_max(const int* __restrict__ ei, const float* __restrict__ a_src,
              const float* __restrict__ a_dst, float* __restrict__ alpha,
              unsigned* __restrict__ m_bits, int E, int T) {
    int e = blockIdx.x * blockDim.x + threadIdx.x;
    if (e >= T) return;
    int src, dst;
    if (e < E) { src = ei[e]; dst = ei[E + e]; }
    else       { src = dst = e - E; }
    float sc = a_src[src] + a_dst[dst];
    sc = (sc > 0.0f) ? sc : NEG_SLOPE * sc;
    alpha[e] = sc;
    atomicMax(&m_bits[dst], fkey(sc));
}

// ---------------------------------------------------------------------------
// Pass B: exp(score - max) -> alpha slot; segment-sum via float atomicAdd.
// ---------------------------------------------------------------------------
__global__ void __launch_bounds__(256)
gat_exp_sum(const int* __restrict__ ei, float* __restrict__ alpha,
            const unsigned* __restrict__ m_bits, float* __restrict__ s,
            int E, int T) {
    int e = blockIdx.x * blockDim.x + threadIdx.x;
    if (e >= T) return;
    int dst = (e < E) ? ei[E + e] : (e - E);
    float ex = __expf(alpha[e] - funkey(m_bits[dst]));
    alpha[e] = ex;
    atomicAdd(&s[dst], ex);
}

// ---------------------------------------------------------------------------
// Pass C1: alpha /= s[dst] in place (separate pass: no read/write race).
// ---------------------------------------------------------------------------
__global__ void __launch_bounds__(256)
gat_norm(const int* __restrict__ ei, float* __restrict__ alpha,
         const float* __restrict__ s, int E, int T) {
    int e = blockIdx.x * blockDim.x + threadIdx.x;
    if (e >= T) return;
    int dst = (e < E) ? ei[E + e] : (e - E);
    alpha[e] = alpha[e] / s[dst];
}

// ---------------------------------------------------------------------------
// Pass C2: out[dst,:] += alpha * h[src,:]; 64 threads per edge, channels
// coalesced; out and h are L2-resident (25.6 MB each, 192 MB L2).
// ---------------------------------------------------------------------------
__global__ void __launch_bounds__(256)
gat_scatter(const int* __restrict__ ei, const float* __restrict__ alpha,
            const float* __restrict__ h, float* __restrict__ out,
            int E, long long T) {
    long long gid = (long long)blockIdx.x * 256 + threadIdx.x;
    long long e = gid >> 6;
    if (e >= T) return;
    int c = (int)(gid & 63);
    int src, dst;
    if (e < (long long)E) { src = ei[e]; dst = ei[(long long)E + e]; }
    else                  { src = dst = (int)(e - E); }
    float a = alpha[e];
    atomicAdd(&out[(size_t)dst * HID + c], a * h[(size_t)src * HID + c]);
}

// ---------------------------------------------------------------------------
extern "C" void kernel_launch(void* const* d_in, const int* in_sizes, int n_in,
                              void* d_out, int out_size, void* d_ws, size_t ws_size,
                              hipStream_t stream) {
    const float* x       = (const float*)d_in[0];
    const int*   ei      = (const int*)d_in[1];
    const float* W       = (const float*)d_in[2];
    const float* att_src = (const float*)d_in[3];
    const float* att_dst = (const float*)d_in[4];
    const float* bias    = (const float*)d_in[5];

    const int N = in_sizes[0] / IN_CH;   // 100000
    const int E = in_sizes[1] / 2;       // 3200000
    const int T = E + N;                 // with self loops

    float* out   = (float*)d_out;                 // [N*HID]
    float* alpha = out + (size_t)N * HID;         // [T]

    char* ws = (char*)d_ws;
    float*    h      = (float*)ws;    ws += (size_t)N * HID * sizeof(float);
    float*    a_src  = (float*)ws;    ws += (size_t)N * sizeof(float);
    float*    a_dst  = (float*)ws;    ws += (size_t)N * sizeof(float);
    unsigned* m_bits = (unsigned*)ws; ws += (size_t)N * sizeof(unsigned);
    float*    sden   = (float*)ws;

    // 1) init out=bias, m=-inf key, s=0
    gat_init<<<(N * HID + 255) / 256, 256, 0, stream>>>(out, bias, m_bits, sden, N);

    // 2) h = x @ W  (WMMA f32 16x16x4, 8 row-tiles per 256-thread block)
    int rowTiles = (N + 15) / 16;
    gat_gemm_wmma<<<(rowTiles + 7) / 8, 256, 0, stream>>>(x, W, h, N);

    // 3) per-node attention halves
    gat_att_dots<<<(N + 255) / 256, 256, 0, stream>>>(h, att_src, att_dst, a_src, a_dst, N);

    // 4) scores + segment max
    gat_score_max<<<(T + 255) / 256, 256, 0, stream>>>(ei, a_src, a_dst, alpha, m_bits, E, T);

    // 5) exp + segment sum
    gat_exp_sum<<<(T + 255) / 256, 256, 0, stream>>>(ei, alpha, m_bits, sden, E, T);

    // 6) normalize alpha
    gat_norm<<<(T + 255) / 256, 256, 0, stream>>>(ei, alpha, sden, E, T);

    // 7) weighted scatter-aggregate
    long long work = (long long)T * HID;
    gat_scatter<<<(unsigned)((work + 255) / 256), 256, 0, stream>>>(ei, alpha, h, out, E, (long long)T);
}